// GFlowCayleyLinear_9723805958251
// MI455X (gfx1250) — compile-verified
//
#include <hip/hip_runtime.h>

typedef float v2f __attribute__((ext_vector_type(2)));
typedef float v8f __attribute__((ext_vector_type(8)));

#define B_DIM  1024
#define T_DIM  64
#define A_DIM  6
#define E_DIM  48
#define H_DIM  64
#define NPAIR  (B_DIM * T_DIM)      /* 65536 (b,t) pairs */
#define TILE_M 16
#define WAVES_PER_BLOCK 4
#define TPB    (WAVES_PER_BLOCK * 32)

/* LDS row paddings: keep DS accesses 8B/16B aligned and spread across banks */
#define XPAD   52   /* X tile rows: 208B stride (16B aligned for b128 stores)   */
#define HPAD   66   /* H tile rows: 264B stride (8B aligned for b64 frag loads) */
#define W1TPAD 50   /* transposed W1: [n][k], 200B stride, b64-aligned          */
#define W2TPAD 66   /* transposed W2: [n][k], 264B stride, b64-aligned          */

__device__ __forceinline__ float softplusf(float x) {
    /* numerically stable: log1p(exp(-|x|)) + max(x,0) (matches jax.nn.softplus) */
    return log1pf(expf(-fabsf(x))) + fmaxf(x, 0.0f);
}

__global__ __launch_bounds__(TPB)
void gfn_flow_wmma(const float* __restrict__ fwd,
                   const float* __restrict__ bwd,
                   const float* __restrict__ W1,
                   const float* __restrict__ b1,
                   const float* __restrict__ W2,
                   const float* __restrict__ b2,
                   float* __restrict__ foutG,
                   float* __restrict__ finG)
{
    /* transposed weights: B fragment (K=k0,k0+1 at column n) = one ds_load_b64 */
    __shared__ float sW1T[H_DIM][W1TPAD];   /* sW1T[n][k] = W1[k*64 + n]        */
    __shared__ float sW2T[16][W2TPAD];      /* sW2T[n][k] = W2[k*6 + n], 0-pad  */
    __shared__ float sB1[H_DIM];
    __shared__ float sB2[8];
    __shared__ float sX[WAVES_PER_BLOCK][TILE_M][XPAD];
    __shared__ float sH[WAVES_PER_BLOCK][TILE_M][HPAD];
    __shared__ float sG[WAVES_PER_BLOCK][TILE_M][8];

    const int tid = threadIdx.x;
    /* cooperative one-time (per block) transposed weight load; W1/W2 are tiny
       and L2-resident, so the uncoalesced transpose reads are negligible */
    for (int i = tid; i < E_DIM * H_DIM; i += TPB) {
        const int n = i / E_DIM, k = i - n * E_DIM;
        sW1T[n][k] = W1[k * H_DIM + n];
    }
    for (int i = tid; i < 16 * H_DIM; i += TPB) {
        const int n = i >> 6, k = i & 63;
        sW2T[n][k] = (n < A_DIM) ? W2[k * A_DIM + n] : 0.0f;
    }
    if (tid < H_DIM) sB1[tid] = b1[tid];
    if (tid < 8)     sB2[tid] = (tid < A_DIM) ? b2[tid] : 0.0f;
    __syncthreads();

    const int wave  = tid >> 5;
    const int lane  = tid & 31;
    const int tile  = blockIdx.x * WAVES_PER_BLOCK + wave;
    const int pair0 = tile * TILE_M;

    /* f32 WMMA 16x16x4 fragment mapping (ISA 7.12.2):
       lanes 0-15 hold K = 4kb+{0,1}; lanes 16-31 hold K = 4kb+{2,3} */
    const int hihalf = lane >> 4;
    const int lrow   = lane & 15;
    const int koff   = hihalf * 2;

    /* slot-invariant W2 B-fragments: keep the serial K=64 chain fed from regs */
    v2f w2f[16];
    #pragma unroll
    for (int kb = 0; kb < 16; ++kb)
        w2f[kb] = *(const v2f*)&sW2T[lrow][4 * kb + koff];

    const float bb2 = sB2[lrow];

    float finAcc  = 0.0f;
    float foutVal = 0.0f;

    const int xrow  = lane >> 1;
    const int cbase = (lane & 1) * 24;

    for (int slot = 0; slot < 7; ++slot) {
        /* ---- stage X tile (16 rows x 48 f32) into LDS, b128 each ---- */
        {
            const float* src = (slot == 0) ? fwd : bwd;
            const float* rp  = src + ((size_t)(pair0 + xrow) * 7 + slot) * E_DIM + cbase;
            #pragma unroll
            for (int i = 0; i < 6; ++i) {
                const float4 v = *(const float4*)(rp + 4 * i);
                *(float4*)&sX[wave][xrow][cbase + 4 * i] = v;
            }
            /* prefetch next slot's rows (global_prefetch_b8) */
            if (slot < 6) {
                const float* srcn = bwd;  /* slots 1..6 all come from bwd */
                const float* rpn  = srcn + ((size_t)(pair0 + xrow) * 7 + slot + 1) * E_DIM + cbase;
                __builtin_prefetch(rpn, 0, 0);
            }
        }

        /* ---- A fragments for X (12 K-blocks of 4) ---- */
        v2f afr[12];
        #pragma unroll
        for (int kb = 0; kb < 12; ++kb)
            afr[kb] = *(const v2f*)&sX[wave][lrow][4 * kb + koff];

        /* ---- H = relu(X @ W1 + b1): bias enters through C ---- */
        #pragma unroll
        for (int nt = 0; nt < 4; ++nt) {
            const int   col = nt * 16 + lrow;
            const float bb  = sB1[col];
            v8f acc = {bb, bb, bb, bb, bb, bb, bb, bb};
            #pragma unroll
            for (int kb = 0; kb < 12; ++kb) {
                const v2f bfr = *(const v2f*)&sW1T[col][4 * kb + koff];
                acc = __builtin_amdgcn_wmma_f32_16x16x4_f32(
                    false, afr[kb], false, bfr, (short)0, acc, false, false);
            }
            #pragma unroll
            for (int j = 0; j < 8; ++j) {
                /* C/D layout: element (m = j + 8*hihalf, n = col) */
                sH[wave][j + 8 * hihalf][col] = fmaxf(acc[j], 0.0f);
            }
        }

        /* ---- G = H @ W2p + b2: one 16x16 tile, K = 64 -> 16 WMMAs ---- */
        v8f g = {bb2, bb2, bb2, bb2, bb2, bb2, bb2, bb2};
        #pragma unroll
        for (int kb = 0; kb < 16; ++kb) {
            const v2f a2 = *(const v2f*)&sH[wave][lrow][4 * kb + koff];
            g = __builtin_amdgcn_wmma_f32_16x16x4_f32(
                false, a2, false, w2f[kb], (short)0, g, false, false);
        }

        /* ---- stash the 6 valid columns, then per-row reduce (lanes 0-15) */
        if (lrow < A_DIM) {
            #pragma unroll
            for (int j = 0; j < 8; ++j)
                sG[wave][j + 8 * hihalf][lrow] = g[j];
        }
        if (lane < 16) {
            if (slot == 0) {
                float s = 0.0f;
                #pragma unroll
                for (int a = 0; a < A_DIM; ++a)
                    s += softplusf(sG[wave][lane][a]);
                foutVal = s;                          /* Fout = sum_a F[...,a] */
            } else {
                /* diagonal: backward slot s uses only head (s-1) */
                finAcc += softplusf(sG[wave][lane][slot - 1]);
            }
        }
    }

    if (lane < 16) {
        foutG[pair0 + lane] = foutVal;
        finG [pair0 + lane] = finAcc;
    }
}

/* one thread per b: exclusive cumsum over T and the 6-channel stack */
__global__ __launch_bounds__(256)
void gfn_finalize(const float* __restrict__ foutG,
                  const float* __restrict__ finG,
                  const float* __restrict__ pinit,
                  const float* __restrict__ prew,
                  const float* __restrict__ initial_flow,
                  float* __restrict__ out)
{
    const int b = blockIdx.x * blockDim.x + threadIdx.x;
    if (b >= B_DIM) return;
    const float ifl = expf(initial_flow[0]);   /* REF_INITFLOW = REWARD_RESCALE = 1 */
    float run = 0.0f;
    for (int t = 0; t < T_DIM; ++t) {
        const int   i  = b * T_DIM + t;
        const float fo = foutG[i];
        const float fi = finG[i];
        const float r  = prew[i];
        float* o = out + (size_t)i * 6;
        o[0] = fi;                 /* Fin   */
        o[1] = fo;                 /* Fout  */
        o[2] = r;                  /* R     */
        o[3] = pinit[i] * ifl;     /* Finit */
        o[4] = run;                /* p (exclusive cumsum) */
        o[5] = r;                  /* paths_reward */
        run += logf(1e-20f + fo) - logf(1e-20f + fo + r);
    }
}

extern "C" void kernel_launch(void* const* d_in, const int* in_sizes, int n_in,
                              void* d_out, int out_size, void* d_ws, size_t ws_size,
                              hipStream_t stream) {
    const float* fwd          = (const float*)d_in[0];
    const float* bwd          = (const float*)d_in[1];
    const float* pinit        = (const float*)d_in[2];
    const float* prew         = (const float*)d_in[3];
    const float* W1           = (const float*)d_in[4];
    const float* b1           = (const float*)d_in[5];
    const float* W2           = (const float*)d_in[6];
    const float* b2           = (const float*)d_in[7];
    const float* initial_flow = (const float*)d_in[8];
    float* out = (float*)d_out;

    float* foutG = (float*)d_ws;           /* NPAIR floats */
    float* finG  = foutG + NPAIR;          /* NPAIR floats */

    const int tiles  = NPAIR / TILE_M;            /* 4096 */
    const int blocks = tiles / WAVES_PER_BLOCK;   /* 1024 */
    hipLaunchKernelGGL(gfn_flow_wmma, dim3(blocks), dim3(TPB), 0, stream,
                       fwd, bwd, W1, b1, W2, b2, foutG, finG);
    hipLaunchKernelGGL(gfn_finalize, dim3((B_DIM + 255) / 256), dim3(256), 0, stream,
                       foutG, finG, pinit, prew, initial_flow, out);
}